// Attention_CA_12111807775305
// MI455X (gfx1250) — compile-verified
//
#include <hip/hip_runtime.h>
#include <math.h>

// Problem constants (from reference shapes)
#define B      2
#define C      384
#define HW     65536
#define L      4096
#define TILES  16      // HW / L
#define HEADS  8
#define CQ     48      // C / HEADS
#define CK     12      // (C/4) / HEADS
#define CC4    96      // C / 4

typedef __attribute__((ext_vector_type(2))) float v2f;
typedef __attribute__((ext_vector_type(8))) float v8f;

// ---------------------------------------------------------------------------
// Kernel 1: per Q row (65536 elems): fused block-sum (16 x 4096 -> 4096) and
// sum-of-squares for L2 normalization. One block per (b,c) row.
// ---------------------------------------------------------------------------
__global__ __launch_bounds__(256) void k_qreduce(const float* __restrict__ Q,
                                                 float* __restrict__ qs,
                                                 float* __restrict__ invq) {
  const int row = blockIdx.x;                 // b*C + c
  const int tid = threadIdx.x;
  const float* __restrict__ qrow = Q + (size_t)row * HW;
  float ss = 0.f;
#pragma unroll
  for (int chunk = 0; chunk < 4; ++chunk) {
    const int col = chunk * 1024 + tid * 4;
    float4 acc = make_float4(0.f, 0.f, 0.f, 0.f);
#pragma unroll
    for (int t = 0; t < TILES; ++t) {
      const float4 v = *(const float4*)(qrow + (size_t)t * L + col);
      acc.x += v.x; acc.y += v.y; acc.z += v.z; acc.w += v.w;
      ss += v.x * v.x + v.y * v.y + v.z * v.z + v.w * v.w;
    }
    *(float4*)(qs + (size_t)row * L + col) = acc;
  }
  // block reduce ss (8 waves of 32)
  for (int off = 16; off > 0; off >>= 1) ss += __shfl_down(ss, off, 32);
  __shared__ float wsum[8];
  if ((tid & 31) == 0) wsum[tid >> 5] = ss;
  __syncthreads();
  if (tid == 0) {
    float t = 0.f;
    for (int i = 0; i < 8; ++i) t += wsum[i];
    invq[row] = 1.0f / fmaxf(sqrtf(t), 1e-12f);
  }
}

// ---------------------------------------------------------------------------
// Kernel 2: avg_pool1d(kernel=4) over kv channel dim + L2-normalize each
// pooled row of length 4096. One block per (b, pooled-channel).
// ---------------------------------------------------------------------------
__global__ __launch_bounds__(1024) void k_poolnorm(const float* __restrict__ kv,
                                                   float* __restrict__ kn) {
  const int bj = blockIdx.x;                  // b*CC4 + j
  const int b = bj / CC4, j = bj % CC4;
  const int tid = threadIdx.x;
  const int col = tid * 4;
  const float* base = kv + ((size_t)(b * C + j * 4)) * L + col;
  float4 p = make_float4(0.f, 0.f, 0.f, 0.f);
#pragma unroll
  for (int r = 0; r < 4; ++r) {
    const float4 v = *(const float4*)(base + (size_t)r * L);
    p.x += v.x; p.y += v.y; p.z += v.z; p.w += v.w;
  }
  p.x *= 0.25f; p.y *= 0.25f; p.z *= 0.25f; p.w *= 0.25f;
  float ss = p.x * p.x + p.y * p.y + p.z * p.z + p.w * p.w;
  for (int off = 16; off > 0; off >>= 1) ss += __shfl_down(ss, off, 32);
  __shared__ float wsum[32];
  __shared__ float s_inv;
  if ((tid & 31) == 0) wsum[tid >> 5] = ss;
  __syncthreads();
  if (tid == 0) {
    float t = 0.f;
    for (int i = 0; i < 32; ++i) t += wsum[i];
    s_inv = 1.0f / fmaxf(sqrtf(t), 1e-12f);
  }
  __syncthreads();
  const float inv = s_inv;
  p.x *= inv; p.y *= inv; p.z *= inv; p.w *= inv;
  *(float4*)(kn + (size_t)bj * L + col) = p;
}

// ---------------------------------------------------------------------------
// Kernel 3a: attn logits + softmax. One block per q row: 12 dots of length
// 4096 against the head's normalized K rows; then softmax over ck=12.
// ---------------------------------------------------------------------------
__global__ __launch_bounds__(256) void k_attn(const float* __restrict__ qs,
                                              const float* __restrict__ invq,
                                              const float* __restrict__ kn,
                                              float* __restrict__ attn) {
  const int row = blockIdx.x;                 // b*C + c
  const int b = row / C, c = row % C;
  const int head = c / CQ;
  const int tid = threadIdx.x;
  const float* __restrict__ kh = kn + ((size_t)(b * CC4 + head * CK)) * L;
  const float* __restrict__ qr = qs + (size_t)row * L;
  float acc[CK];
#pragma unroll
  for (int k = 0; k < CK; ++k) acc[k] = 0.f;
  for (int j = tid; j < L; j += 256) {
    const float qv = qr[j];
#pragma unroll
    for (int k = 0; k < CK; ++k) acc[k] += qv * kh[(size_t)k * L + j];
  }
#pragma unroll
  for (int k = 0; k < CK; ++k)
    for (int off = 16; off > 0; off >>= 1) acc[k] += __shfl_down(acc[k], off, 32);
  __shared__ float red[CK];
  if (tid < CK) red[tid] = 0.f;
  __syncthreads();
  if ((tid & 31) == 0) {
#pragma unroll
    for (int k = 0; k < CK; ++k) atomicAdd(&red[k], acc[k]);
  }
  __syncthreads();
  if (tid == 0) {
    const float iq = invq[row];
    float logit[CK], m = -3.0e38f;
#pragma unroll
    for (int k = 0; k < CK; ++k) { logit[k] = red[k] * iq; m = fmaxf(m, logit[k]); }
    float s = 0.f;
#pragma unroll
    for (int k = 0; k < CK; ++k) { logit[k] = __expf(logit[k] - m); s += logit[k]; }
    const float is = 1.0f / s;
    float* arow = attn + (size_t)row * CK;
#pragma unroll
    for (int k = 0; k < CK; ++k) arow[k] = logit[k] * is;
  }
}

// ---------------------------------------------------------------------------
// Kernel 3b: out_small = attn(48x12) @ kn(12x4096) per (b,head). One block
// per (b,head, 1024-col chunk); kn chunk cached in registers (12 x float4).
// ---------------------------------------------------------------------------
__global__ __launch_bounds__(256) void k_outsmall(const float* __restrict__ attn,
                                                  const float* __restrict__ kn,
                                                  float* __restrict__ outs) {
  const int bid = blockIdx.x;                 // (b*HEADS + head)*4 + chunk
  const int chunk = bid & 3;
  const int bh = bid >> 2;
  const int b = bh / HEADS, head = bh % HEADS;
  const int tid = threadIdx.x;
  __shared__ float a_s[CQ * CK];
  for (int i = tid; i < CQ * CK; i += 256)
    a_s[i] = attn[((size_t)(b * C + head * CQ)) * CK + i];
  __syncthreads();
  const int col = chunk * 1024 + tid * 4;
  const float* __restrict__ kh = kn + ((size_t)(b * CC4 + head * CK)) * L + col;
  float4 kr[CK];
#pragma unroll
  for (int k = 0; k < CK; ++k) kr[k] = *(const float4*)(kh + (size_t)k * L);
  float* orow = outs + ((size_t)(b * C + head * CQ)) * L + col;
  for (int c = 0; c < CQ; ++c) {
    float4 o = make_float4(0.f, 0.f, 0.f, 0.f);
#pragma unroll
    for (int k = 0; k < CK; ++k) {
      const float a = a_s[c * CK + k];
      o.x += a * kr[k].x; o.y += a * kr[k].y; o.z += a * kr[k].z; o.w += a * kr[k].w;
    }
    *(float4*)(orow + (size_t)c * L) = o;
  }
}

// ---------------------------------------------------------------------------
// Kernel 4: projection fin = W(384x384) @ out_small(384x4096), fp32 WMMA
// (V_WMMA_F32_16X16X4_F32). One wave per 16x16 tile; K-loop of 96 WMMAs.
// Each lane writes its D column to all 16 periodic positions of d_out.
// A layout (16x4 f32): lanes0-15 M=lane, v0=K0,v1=K1; lanes16-31 v0=K2,v1=K3.
// B layout (4x16 f32): lanes0-15 N=lane, v0=K0,v1=K1; lanes16-31 v0=K2,v1=K3.
// D layout: vgpr j -> M=j (lanes<16) / M=8+j (lanes>=16), N=lane&15.
// ---------------------------------------------------------------------------
__global__ __launch_bounds__(256) void k_proj(const float* __restrict__ W,
                                              const float* __restrict__ outs,
                                              float* __restrict__ out) {
  const int bid = blockIdx.x;
  const int b = bid / 768;                    // 768 = 256 ntiles * 3 mgroups
  const int rem = bid % 768;
  const int ntile = rem / 3;
  const int mgroup = rem % 3;
  const int tid = threadIdx.x;
  const int wave = tid >> 5;
  const int lane = tid & 31;
  const int mtile = mgroup * 8 + wave;        // 0..23
  const int o0 = mtile * 16;
  const int m0 = ntile * 16;
  const int lhalf = lane >> 4;                // 0 or 1
  const int l15 = lane & 15;
  const int arow = o0 + l15;
  const int koff = lhalf * 2;
  const int bcol = m0 + l15;
  const float* __restrict__ S = outs + (size_t)b * C * L;

  v8f acc = {};
  for (int k0 = 0; k0 < C; k0 += 4) {
    v2f a, bb;
    const float2 aw = *(const float2*)(W + (size_t)arow * C + k0 + koff);
    a.x = aw.x; a.y = aw.y;
    bb.x = S[(size_t)(k0 + koff) * L + bcol];
    bb.y = S[(size_t)(k0 + koff + 1) * L + bcol];
    acc = __builtin_amdgcn_wmma_f32_16x16x4_f32(
        /*neg_a=*/false, a, /*neg_b=*/false, bb,
        /*c_mod=*/(short)0, acc, /*reuse_a=*/false, /*reuse_b=*/false);
  }

  const int rbase = o0 + lhalf * 8;
  float* obase = out + (size_t)b * C * (size_t)HW;
#pragma unroll
  for (int j = 0; j < 8; ++j) {
    const float val = acc[j];
    float* p = obase + (size_t)(rbase + j) * HW + bcol;
#pragma unroll
    for (int t = 0; t < TILES; ++t) p[t * L] = val;
  }
}

// ---------------------------------------------------------------------------
extern "C" void kernel_launch(void* const* d_in, const int* in_sizes, int n_in,
                              void* d_out, int out_size, void* d_ws, size_t ws_size,
                              hipStream_t stream) {
  const float* Q  = (const float*)d_in[0];   // [2,384,256,256]
  const float* kv = (const float*)d_in[1];   // [2,384,4096]
  const float* W  = (const float*)d_in[2];   // [384,384]
  float* out = (float*)d_out;                // [2,384,256,256]
  float* ws = (float*)d_ws;

  // workspace layout (floats), total ~27.7 MB
  float* qs   = ws;                               // B*C*L
  float* invq = qs   + (size_t)B * C * L;         // B*C
  float* kn   = invq + (size_t)B * C;             // B*CC4*L
  float* attn = kn   + (size_t)B * CC4 * L;       // B*C*CK
  float* outs = attn + (size_t)B * C * CK;        // B*C*L

  k_qreduce <<<B * C,         256,  0, stream>>>(Q, qs, invq);
  k_poolnorm<<<B * CC4,       1024, 0, stream>>>(kv, kn);
  k_attn    <<<B * C,         256,  0, stream>>>(qs, invq, kn, attn);
  k_outsmall<<<B * HEADS * 4, 256,  0, stream>>>(attn, kn, outs);
  k_proj    <<<B * 256 * 3,   256,  0, stream>>>(W, outs, out);
}